// RITS_41420664602661
// MI455X (gfx1250) — compile-verified
//
#include <hip/hip_runtime.h>
#include <hip/hip_bf16.h>
#include <math.h>

// ---------------- problem constants ----------------
#define BB 256   // batch
#define TT 100   // time steps
#define FF 256   // features
#define HH 512   // hidden
#define TF (TT * FF)        // row stride of [B,T,F] tensors
#define BTF ((size_t)BB * TT * FF)

// ---------------- workspace layout (floats) ----------------
#define OFF_H    0
#define OFF_C    (OFF_H + BB * HH)              // 131072
#define OFF_HDEC (OFF_C + BB * HH)              // 262144
#define OFF_GM   (OFF_HDEC + BB * HH)           // 393216  [gamma_x | m]
#define OFF_XH   (OFF_GM + BB * 2 * FF)         // 524288
#define OFF_XC   (OFF_XH + BB * FF)             // 589824
#define OFF_INP  (OFF_XC + BB * FF)             // 655360  [c_c | m]
#define OFF_LN   (OFF_INP + BB * 2 * FF)        // 786432  (3*T used)
#define OFF_MS   (OFF_LN + 384)                 // 786816  (T used)

typedef __attribute__((ext_vector_type(2))) float v2f;
typedef __attribute__((ext_vector_type(8))) float v8f;

__device__ __forceinline__ v2f ldv2(const float* p) { return *(const v2f*)p; }

// CDNA5 native fp32 WMMA: D(16x16,f32) = A(16x4,f32) x B(4x16,f32) + C
__device__ __forceinline__ v8f wmma4(v2f a, v2f b, v8f c) {
    return __builtin_amdgcn_wmma_f32_16x16x4_f32(false, a, false, b,
                                                 (short)0, c, false, false);
}

__device__ __forceinline__ float wave_sum(float v) {
    for (int off = 16; off > 0; off >>= 1) v += __shfl_down(v, off, 32);
    return v;
}

__device__ __forceinline__ float sigf(float x) { return 1.0f / (1.0f + expf(-x)); }

// ======================================================================
// init: zero h_tmp, c, loss numerators, mask sums
// ======================================================================
__global__ void k_zero(float* ws) {
    int i = blockIdx.x * blockDim.x + threadIdx.x;
    if (i < 2 * BB * HH) ws[OFF_H + i] = 0.0f;
    else                 ws[OFF_LN + (i - 2 * BB * HH)] = 0.0f;
}

// ======================================================================
// per-step mask sums: ms[t] = sum over [B,F] of masks[:,t,:]
// ======================================================================
__global__ void k_msum(const float* __restrict__ masks, float* ws) {
    int t = blockIdx.x;
    float s = 0.0f;
    for (int i = threadIdx.x; i < BB * FF; i += blockDim.x) {
        int b = i >> 8, f = i & 255;
        s += masks[(size_t)b * TF + t * FF + f];
    }
    s = wave_sum(s);
    if ((threadIdx.x & 31) == 0) atomicAdd(&ws[OFF_MS + t], s);
}

// ======================================================================
// K1: gamma_h GEMM (M=256 N=512 K=256) + h_dec = h_tmp * gamma_h.
//     blockIdx.y == 8 slice does the gamma_x elementwise work instead
//     (fused former k_gx kernel; branches out before any WMMA).
// grid(4, 9), block 128
// ======================================================================
__global__ void __launch_bounds__(128) k_gamma_h(
        const float* __restrict__ deltas, const float* __restrict__ masks,
        const float* __restrict__ Wth, const float* __restrict__ bth,
        const float* __restrict__ Wtx, const float* __restrict__ btx,
        float* __restrict__ ws, int t) {
    if (blockIdx.y == 8) {
        // elementwise gamma_x + stage mask into gm[:,F:] and inp[:,F:]
        // rows [blockIdx.x*64, +64), 64*256 elems over 128 threads
        int base = blockIdx.x * 64;
        for (int idx = threadIdx.x; idx < 64 * FF; idx += 128) {
            int b = base + (idx >> 8);
            int f = idx & 255;
            float d = deltas[(size_t)b * TF + t * FF + f];
            float m = masks [(size_t)b * TF + t * FF + f];
            float g = expf(-fmaxf(d * Wtx[f * FF + f] + btx[f], 0.0f));
            ws[OFF_GM  + b * (2 * FF) + f] = g;
            ws[OFF_GM  + b * (2 * FF) + FF + f] = m;
            ws[OFF_INP + b * (2 * FF) + FF + f] = m;
        }
        return;
    }
    const int lane = threadIdx.x & 31, wv = threadIdx.x >> 5;
    const int lo = lane & 15, hi = lane >> 4;
    const int m0 = blockIdx.x * 64 + wv * 16;
    const int nb = blockIdx.y * 64;
    const float* A = deltas + (size_t)t * FF;   // + row * TF

    v8f z = {0.f,0.f,0.f,0.f,0.f,0.f,0.f,0.f};
    v8f acc[4] = {z, z, z, z};
    for (int k = 0; k < FF; k += 4) {
        v2f a = ldv2(A + (size_t)(m0 + lo) * TF + k + 2 * hi);
#pragma unroll
        for (int j = 0; j < 4; ++j) {
            int n = nb + j * 16 + lo;
            v2f b = ldv2(Wth + (size_t)n * FF + k + 2 * hi);
            acc[j] = wmma4(a, b, acc[j]);
        }
    }
    float* h_tmp = ws + OFF_H;
    float* h_dec = ws + OFF_HDEC;
#pragma unroll
    for (int j = 0; j < 4; ++j) {
        int col = nb + j * 16 + lo;
        float bb = bth[col];
#pragma unroll
        for (int r = 0; r < 8; ++r) {
            int row = m0 + r + 8 * hi;
            float g = expf(-fmaxf(acc[j][r] + bb, 0.0f));
            h_dec[row * HH + col] = h_tmp[row * HH + col] * g;
        }
    }
}

// ======================================================================
// K2: x_h = h_dec @ W_hist^T + b ; x_c = m*x+(1-m)*x_h ; loss1
// GEMM M=256 N=256 K=512.  grid(4,4), block 128
// ======================================================================
__global__ void __launch_bounds__(128) k_xh(
        const float* __restrict__ values, const float* __restrict__ masks,
        const float* __restrict__ Wh, const float* __restrict__ bh,
        float* __restrict__ ws, int t) {
    const int lane = threadIdx.x & 31, wv = threadIdx.x >> 5;
    const int lo = lane & 15, hi = lane >> 4;
    const int m0 = blockIdx.x * 64 + wv * 16;
    const int nb = blockIdx.y * 64;
    const float* A = ws + OFF_HDEC;

    v8f z = {0.f,0.f,0.f,0.f,0.f,0.f,0.f,0.f};
    v8f acc[4] = {z, z, z, z};
    for (int k = 0; k < HH; k += 4) {
        v2f a = ldv2(A + (size_t)(m0 + lo) * HH + k + 2 * hi);
#pragma unroll
        for (int j = 0; j < 4; ++j) {
            int n = nb + j * 16 + lo;
            v2f b = ldv2(Wh + (size_t)n * HH + k + 2 * hi);
            acc[j] = wmma4(a, b, acc[j]);
        }
    }
    float* xhb = ws + OFF_XH;
    float* xcb = ws + OFF_XC;
    float l1 = 0.0f;
#pragma unroll
    for (int j = 0; j < 4; ++j) {
        int col = nb + j * 16 + lo;
        float bb = bh[col];
#pragma unroll
        for (int r = 0; r < 8; ++r) {
            int row = m0 + r + 8 * hi;
            float xh = acc[j][r] + bb;
            float x  = values[(size_t)row * TF + t * FF + col];
            float mk = masks [(size_t)row * TF + t * FF + col];
            xhb[row * FF + col] = xh;
            xcb[row * FF + col] = mk * x + (1.0f - mk) * xh;
            l1 += fabsf(xh - x) * mk;
        }
    }
    l1 = wave_sum(l1);
    if (lane == 0) atomicAdd(&ws[OFF_LN + 3 * t + 0], l1);
}

// ======================================================================
// K3: fused z_h (diag-masked W_feat) + alpha (W_comb over [gamma_x|m])
//     -> c_h, c_c (into inp[:, :F]), estimation store, loss2, loss3
// ======================================================================
__global__ void __launch_bounds__(128) k_fuse(
        const float* __restrict__ values, const float* __restrict__ masks,
        const float* __restrict__ Wf, const float* __restrict__ bf,
        const float* __restrict__ Wc, const float* __restrict__ bc,
        float* __restrict__ ws, float* __restrict__ out, int t) {
    const int lane = threadIdx.x & 31, wv = threadIdx.x >> 5;
    const int lo = lane & 15, hi = lane >> 4;
    const int m0 = blockIdx.x * 64 + wv * 16;
    const int nb = blockIdx.y * 64;

    v8f z = {0.f,0.f,0.f,0.f,0.f,0.f,0.f,0.f};
    v8f accz[4] = {z, z, z, z};
    v8f acca[4] = {z, z, z, z};

    // z_h: K = F over x_c and W_feat with zeroed diagonal
    const float* Axc = ws + OFF_XC;
    for (int k = 0; k < FF; k += 4) {
        v2f a = ldv2(Axc + (size_t)(m0 + lo) * FF + k + 2 * hi);
        int kk = k + 2 * hi;
#pragma unroll
        for (int j = 0; j < 4; ++j) {
            int n = nb + j * 16 + lo;
            v2f b = ldv2(Wf + (size_t)n * FF + kk);
            if (n == kk)     b.x = 0.0f;
            if (n == kk + 1) b.y = 0.0f;
            accz[j] = wmma4(a, b, accz[j]);
        }
    }
    // alpha: K = 2F over gm = [gamma_x | m] and W_comb
    const float* Agm = ws + OFF_GM;
    for (int k = 0; k < 2 * FF; k += 4) {
        v2f a = ldv2(Agm + (size_t)(m0 + lo) * (2 * FF) + k + 2 * hi);
#pragma unroll
        for (int j = 0; j < 4; ++j) {
            int n = nb + j * 16 + lo;
            v2f b = ldv2(Wc + (size_t)n * (2 * FF) + k + 2 * hi);
            acca[j] = wmma4(a, b, acca[j]);
        }
    }
    const float* xhb = ws + OFF_XH;
    float* inp = ws + OFF_INP;
    float l2 = 0.0f, l3 = 0.0f;
#pragma unroll
    for (int j = 0; j < 4; ++j) {
        int col = nb + j * 16 + lo;
        float bfv = bf[col], bcv = bc[col];
#pragma unroll
        for (int r = 0; r < 8; ++r) {
            int row = m0 + r + 8 * hi;
            float zh = fmaxf(accz[j][r] + bfv, 0.0f);
            float al = sigf(acca[j][r] + bcv);
            float xh = xhb[row * FF + col];
            float ch = al * zh + (1.0f - al) * xh;
            float x  = values[(size_t)row * TF + t * FF + col];
            float mk = masks [(size_t)row * TF + t * FF + col];
            inp[row * (2 * FF) + col] = mk * x + (1.0f - mk) * ch;   // c_c
            out[(size_t)row * TF + t * FF + col] = ch;               // estimation
            l2 += fabsf(zh - x) * mk;
            l3 += fabsf(ch - x) * mk;
        }
    }
    l2 = wave_sum(l2);
    l3 = wave_sum(l3);
    if (lane == 0) {
        atomicAdd(&ws[OFF_LN + 3 * t + 1], l2);
        atomicAdd(&ws[OFF_LN + 3 * t + 2], l3);
    }
}

// ======================================================================
// K4: gates = [c_c,m] @ W_ih^T + h_dec @ W_hh^T + b ; LSTM cell update.
// M=256, N=512 per gate (4 gates), K=512+512.  grid(4,16), block 128.
// Weight slabs (128 rows x 32 K) staged cooperatively in LDS: kills the
// 4x per-block redundant global B loads.  Row stride padded to 36 floats
// -> 16-lane ds_load_b64 fragment reads hit 16 distinct banks, and
// 144 B rows keep the b128 fill stores 16 B aligned.
// ======================================================================
#define KS   32
#define KSP  36   // padded LDS row stride (floats)
__global__ void __launch_bounds__(128) k_lstm(
        const float* __restrict__ Wih, const float* __restrict__ Whh,
        const float* __restrict__ bih, const float* __restrict__ bhh,
        float* __restrict__ ws) {
    __shared__ __align__(16) float sw[128 * KSP];   // 18 KB
    const int tid  = threadIdx.x;
    const int lane = tid & 31, wv = tid >> 5;
    const int lo = lane & 15, hi = lane >> 4;
    const int m0 = blockIdx.x * 64 + wv * 16;
    const int nb = blockIdx.y * 32;

    v8f z = {0.f,0.f,0.f,0.f,0.f,0.f,0.f,0.f};
    v8f acc[4][2] = {{z, z}, {z, z}, {z, z}, {z, z}};

#pragma unroll
    for (int half = 0; half < 2; ++half) {
        const float* W = half ? Whh : Wih;                    // ldw = 512 both
        const float* A = ws + (half ? OFF_HDEC : OFF_INP);    // lda = 512 both
        for (int k0 = 0; k0 < HH; k0 += KS) {
            __syncthreads();   // previous slab fully consumed
            // cooperative fill: 128 weight rows x 32 K, b128 per chunk
#pragma unroll
            for (int p = 0; p < 8; ++p) {
                int c    = tid * 8 + p;          // 1024 x 16B chunks
                int row  = c >> 3;               // 0..127
                int part = c & 7;                // 8 x 4 floats per row
                int rg   = (row >> 5) * HH + nb + (row & 31);   // gate*H + col
                *(float4*)(&sw[row * KSP + part * 4]) =
                    *(const float4*)(W + (size_t)rg * (2 * FF) + k0 + part * 4);
            }
            // prefetch next slab into cache while this one is consumed
            if (k0 + KS < HH) {
                int rg = (tid >> 5) * HH + nb + (tid & 31);
                __builtin_prefetch(W + (size_t)rg * (2 * FF) + k0 + KS, 0, 0);
            }
            __syncthreads();   // slab visible to all waves
#pragma unroll
            for (int kk = 0; kk < KS; kk += 4) {
                v2f a = ldv2(A + (size_t)(m0 + lo) * 512 + k0 + kk + 2 * hi);
#pragma unroll
                for (int j = 0; j < 2; ++j) {
#pragma unroll
                    for (int g = 0; g < 4; ++g) {
                        int brow = g * 32 + j * 16 + lo;
                        v2f b = *(const v2f*)(&sw[brow * KSP + kk + 2 * hi]);
                        acc[g][j] = wmma4(a, b, acc[g][j]);
                    }
                }
            }
        }
    }
    float* h_tmp = ws + OFF_H;
    float* cbuf  = ws + OFF_C;
#pragma unroll
    for (int j = 0; j < 2; ++j) {
        int col = nb + j * 16 + lo;
        float bi  = bih[col]           + bhh[col];
        float bfv = bih[HH + col]      + bhh[HH + col];
        float bg  = bih[2 * HH + col]  + bhh[2 * HH + col];
        float bo  = bih[3 * HH + col]  + bhh[3 * HH + col];
#pragma unroll
        for (int r = 0; r < 8; ++r) {
            int row = m0 + r + 8 * hi;
            float ig = acc[0][j][r] + bi;
            float fg = acc[1][j][r] + bfv;
            float gg = acc[2][j][r] + bg;
            float og = acc[3][j][r] + bo;
            float cOld = cbuf[row * HH + col];
            float cNew = sigf(fg) * cOld + sigf(ig) * tanhf(gg);
            float hNew = sigf(og) * tanhf(cNew);
            cbuf[row * HH + col]  = cNew;
            h_tmp[row * HH + col] = hNew;
        }
    }
}

// ======================================================================
// finalize: imputed (in-place on d_out), h copy, recon loss
// ======================================================================
__global__ void k_imputed(const float* __restrict__ values,
                          const float* __restrict__ masks,
                          float* __restrict__ out) {
    size_t i = (size_t)blockIdx.x * blockDim.x + threadIdx.x;
    float m = masks[i];
    out[i] = m * values[i] + (1.0f - m) * out[i];
}

__global__ void k_hcopy(const float* __restrict__ ws, float* __restrict__ out) {
    int i = blockIdx.x * blockDim.x + threadIdx.x;
    out[BTF + i] = ws[OFF_H + i];
}

__global__ void k_loss(const float* __restrict__ ws, float* __restrict__ out) {
    if (threadIdx.x == 0 && blockIdx.x == 0) {
        float l = 0.0f;
        for (int t = 0; t < TT; ++t) {
            float sm = ws[OFF_MS + t] + 1e-9f;
            l += (ws[OFF_LN + 3 * t + 0] + ws[OFF_LN + 3 * t + 1] +
                  ws[OFF_LN + 3 * t + 2]) / sm;
        }
        out[BTF + (size_t)BB * HH] = l / (3.0f * TT);
    }
}

// ======================================================================
extern "C" void kernel_launch(void* const* d_in, const int* in_sizes, int n_in,
                              void* d_out, int out_size, void* d_ws, size_t ws_size,
                              hipStream_t stream) {
    const float* values = (const float*)d_in[0];
    const float* masks  = (const float*)d_in[1];
    const float* deltas = (const float*)d_in[2];
    const float* Wth    = (const float*)d_in[3];
    const float* bth    = (const float*)d_in[4];
    const float* Wtx    = (const float*)d_in[5];
    const float* btx    = (const float*)d_in[6];
    const float* Whist  = (const float*)d_in[7];
    const float* bhist  = (const float*)d_in[8];
    const float* Wfeat  = (const float*)d_in[9];
    const float* bfeat  = (const float*)d_in[10];
    const float* Wcomb  = (const float*)d_in[11];
    const float* bcomb  = (const float*)d_in[12];
    const float* Wih    = (const float*)d_in[13];
    const float* Whh    = (const float*)d_in[14];
    const float* bih    = (const float*)d_in[15];
    const float* bhh    = (const float*)d_in[16];
    float* out = (float*)d_out;
    float* ws  = (float*)d_ws;

    // zero h, c, loss accumulators
    int nz = 2 * BB * HH + 512;
    k_zero<<<(nz + 255) / 256, 256, 0, stream>>>(ws);
    // per-step mask sums
    k_msum<<<TT, 256, 0, stream>>>(masks, ws);

    for (int t = 0; t < TT; ++t) {
        // grid.y == 8 slice carries the fused gamma_x elementwise work
        k_gamma_h<<<dim3(BB / 64, 9), 128, 0, stream>>>(deltas, masks, Wth, bth,
                                                        Wtx, btx, ws, t);
        k_xh<<<dim3(BB / 64, FF / 64), 128, 0, stream>>>(values, masks, Whist, bhist, ws, t);
        k_fuse<<<dim3(BB / 64, FF / 64), 128, 0, stream>>>(values, masks, Wfeat, bfeat,
                                                           Wcomb, bcomb, ws, out, t);
        k_lstm<<<dim3(BB / 64, HH / 32), 128, 0, stream>>>(Wih, Whh, bih, bhh, ws);
    }

    k_imputed<<<(unsigned)(BTF / 256), 256, 0, stream>>>(values, masks, out);
    k_hcopy<<<(BB * HH) / 256, 256, 0, stream>>>(ws, out);
    k_loss<<<1, 32, 0, stream>>>(ws, out);
}